// MemNet_498216206665
// MI455X (gfx1250) — compile-verified
//
#include <hip/hip_runtime.h>
#include <hip/hip_bf16.h>

typedef __attribute__((ext_vector_type(2))) float v2f;
typedef __attribute__((ext_vector_type(8))) float v8f;

#define B_ 64
#define L_ 512
#define T_ 8
#define D_ 1024
#define C_ 3
#define NEG_INF_ (-1e30f)

// ---------------------------------------------------------------------------
// Kernel 1: v_aspect = masked mean of target embeddings -> vec0 [B,D]
// grid = B, block = 256, each thread owns 4 contiguous d via float4
// ---------------------------------------------------------------------------
__global__ __launch_bounds__(256)
void k_vaspect(const int* __restrict__ tgt_ids, const int* __restrict__ tgt_len,
               const float* __restrict__ E, float* __restrict__ vec)
{
    int b = blockIdx.x;
    int d0 = threadIdx.x * 4;
    int tlen = tgt_len[b];
    float4 acc = make_float4(0.f, 0.f, 0.f, 0.f);
    for (int t = 0; t < tlen; ++t) {
        int id = tgt_ids[b * T_ + t];
        const float4 r = *(const float4*)(E + (size_t)id * D_ + d0);
        acc.x += r.x; acc.y += r.y; acc.z += r.z; acc.w += r.w;
    }
    float inv = 1.0f / (float)tlen;
    acc.x *= inv; acc.y *= inv; acc.z *= inv; acc.w *= inv;
    *(float4*)(vec + (size_t)b * D_ + d0) = acc;
}

// ---------------------------------------------------------------------------
// Kernel 2 (hop-invariant): s_mem[b,l] = v_loc[b,l] * dot(E[ctx[b,l]], w_attn[0:D])
// one wave32 per (b,l); 8 waves per block
// ---------------------------------------------------------------------------
__global__ __launch_bounds__(256)
void k_smem(const int* __restrict__ ctx_ids, const int* __restrict__ ctx_len,
            const int* __restrict__ tgt_off, const float* __restrict__ E,
            const float* __restrict__ w_attn, float* __restrict__ s_mem,
            float* __restrict__ vloc)
{
    int gw   = (blockIdx.x * 256 + threadIdx.x) >> 5;   // global wave id in [0, B*L)
    int lane = threadIdx.x & 31;
    int b = gw >> 9;          // /512
    int l = gw & (L_ - 1);
    int id = ctx_ids[b * L_ + l];
    const float* row = E + (size_t)id * D_;
    float p = 0.f;
    #pragma unroll 4
    for (int k = lane; k < D_; k += 32) p += row[k] * w_attn[k];
    #pragma unroll
    for (int off = 16; off > 0; off >>= 1) p += __shfl_xor(p, off, 32);
    int clen = ctx_len[b];
    float vl = 0.f;
    if (l < clen) {
        float dist = fabsf((float)(l - tgt_off[b]));
        vl = 1.f - dist / (float)clen;
    }
    if (lane == 0) {
        s_mem[gw] = vl * p;
        vloc[gw]  = vl;
    }
}

// ---------------------------------------------------------------------------
// Kernel 3 (per hop): s_vec = vec.w_attn[D:2D]+b_attn; scores=tanh(s_mem+s_vec)
// masked softmax over L; emit fused weight w[b,l] = alpha[b,l]*v_loc[b,l]
// grid = B, block = 256
// ---------------------------------------------------------------------------
__global__ __launch_bounds__(256)
void k_softmax(const float* __restrict__ vec, const float* __restrict__ s_mem,
               const float* __restrict__ vloc, const float* __restrict__ w_attn,
               const float* __restrict__ b_attn, const int* __restrict__ ctx_len,
               float* __restrict__ w_out)
{
    __shared__ float red[256];
    int b = blockIdx.x, tid = threadIdx.x;

    // s_vec = vec[b,:] . w_attn[D:2D]
    float part = 0.f;
    #pragma unroll
    for (int d = tid; d < D_; d += 256) part += vec[(size_t)b * D_ + d] * w_attn[D_ + d];
    red[tid] = part; __syncthreads();
    for (int s = 128; s > 0; s >>= 1) { if (tid < s) red[tid] += red[tid + s]; __syncthreads(); }
    float s_vec = red[0] + b_attn[0];
    __syncthreads();

    int clen = ctx_len[b];
    float sc0, sc1;
    {
        int l0 = tid, l1 = tid + 256;
        sc0 = (l0 < clen) ? tanhf(s_mem[b * L_ + l0] + s_vec) : NEG_INF_;
        sc1 = (l1 < clen) ? tanhf(s_mem[b * L_ + l1] + s_vec) : NEG_INF_;
    }
    // max reduce
    red[tid] = fmaxf(sc0, sc1); __syncthreads();
    for (int s = 128; s > 0; s >>= 1) { if (tid < s) red[tid] = fmaxf(red[tid], red[tid + s]); __syncthreads(); }
    float mx = red[0];
    __syncthreads();
    float e0 = __expf(sc0 - mx);
    float e1 = __expf(sc1 - mx);
    red[tid] = e0 + e1; __syncthreads();
    for (int s = 128; s > 0; s >>= 1) { if (tid < s) red[tid] += red[tid + s]; __syncthreads(); }
    float inv_total = 1.0f / red[0];
    w_out[b * L_ + tid]       = e0 * inv_total * vloc[b * L_ + tid];
    w_out[b * L_ + tid + 256] = e1 * inv_total * vloc[b * L_ + tid + 256];
}

// ---------------------------------------------------------------------------
// Kernel 4 (per hop): attn_out[b,d] = sum_l w[b,l] * E[ctx[b,l], d]
// grid = (D/256, B), block = 256; ids+weights staged in LDS,
// next-row prefetch via global_prefetch_b8
// ---------------------------------------------------------------------------
__global__ __launch_bounds__(256)
void k_attn(const int* __restrict__ ctx_ids, const int* __restrict__ ctx_len,
            const float* __restrict__ E, const float* __restrict__ w,
            float* __restrict__ attn_out)
{
    __shared__ float wl[L_];
    __shared__ int   ids[L_];
    int b   = blockIdx.y;
    int tid = threadIdx.x;
    int d   = blockIdx.x * 256 + tid;
    #pragma unroll
    for (int i = 0; i < 2; ++i) {
        int l = tid + i * 256;
        wl[l]  = w[b * L_ + l];
        ids[l] = ctx_ids[b * L_ + l];
    }
    __syncthreads();
    int clen = ctx_len[b];
    float acc = 0.f;
    for (int l = 0; l < clen; ++l) {
        if (l + 1 < clen)
            __builtin_prefetch(E + (size_t)ids[l + 1] * D_ + d, 0, 3);
        acc += wl[l] * E[(size_t)ids[l] * D_ + d];
    }
    attn_out[(size_t)b * D_ + d] = acc;
}

// ---------------------------------------------------------------------------
// Kernel 5 (per hop): lin = vec @ W_lin via V_WMMA_F32_16X16X4_F32, then
// vec_next = lin + attn_out + b_lin.  One wave per 16x16 output tile.
// M=64 -> 4 tiles, N=1024 -> 64 tiles => 256 tiles; 8 waves/block, 32 blocks.
// A-frag (16x4 f32): K = j + 2*(lane>>4), M = lane&15  (ISA 7.12.2)
// B-frag (4x16 f32): K = j + 2*(lane>>4), N = lane&15  (row-striped like C/D)
// C/D (16x16 f32):   M = i + 8*(lane>>4), N = lane&15
// ---------------------------------------------------------------------------
__global__ __launch_bounds__(256)
void k_lin_wmma(const float* __restrict__ vec_in, const float* __restrict__ W_lin,
                const float* __restrict__ b_lin, const float* __restrict__ attn,
                float* __restrict__ vec_out)
{
    int wave = threadIdx.x >> 5;
    int lane = threadIdx.x & 31;
    int tile = blockIdx.x * 8 + wave;   // 0..255
    int m0 = (tile & 3) * 16;
    int n0 = (tile >> 2) * 16;
    int ml   = lane & 15;
    int half = lane >> 4;

    v8f c = {};
    const float* arow = vec_in + (size_t)(m0 + ml) * D_;
    for (int k = 0; k < D_; k += 4) {
        int ka = k + 2 * half;
        v2f a, bf;
        a.x  = arow[ka];
        a.y  = arow[ka + 1];
        bf.x = W_lin[(size_t)ka * D_ + n0 + ml];
        bf.y = W_lin[(size_t)(ka + 1) * D_ + n0 + ml];
        c = __builtin_amdgcn_wmma_f32_16x16x4_f32(
                /*neg_a=*/false, a, /*neg_b=*/false, bf,
                /*c_mod=*/(short)0, c, /*reuse_a=*/false, /*reuse_b=*/false);
    }

    int n = n0 + ml;
    float bias = b_lin[n];
    #pragma unroll
    for (int i = 0; i < 8; ++i) {
        int m = m0 + i + 8 * half;
        size_t off = (size_t)m * D_ + n;
        vec_out[off] = c[i] + attn[off] + bias;
    }
}

// ---------------------------------------------------------------------------
// Kernel 6: logits[b,c] = vec[b,:] . W_out[:,c] + b_out[c]
// grid = B*C, block = 256
// ---------------------------------------------------------------------------
__global__ __launch_bounds__(256)
void k_logits(const float* __restrict__ vec, const float* __restrict__ W_out,
              const float* __restrict__ b_out, float* __restrict__ out)
{
    __shared__ float red[256];
    int b = blockIdx.x / C_;
    int cc = blockIdx.x % C_;
    int tid = threadIdx.x;
    float part = 0.f;
    #pragma unroll
    for (int d = tid; d < D_; d += 256) part += vec[(size_t)b * D_ + d] * W_out[d * C_ + cc];
    red[tid] = part; __syncthreads();
    for (int s = 128; s > 0; s >>= 1) { if (tid < s) red[tid] += red[tid + s]; __syncthreads(); }
    if (tid == 0) out[b * C_ + cc] = red[0] + b_out[cc];
}

// ---------------------------------------------------------------------------
extern "C" void kernel_launch(void* const* d_in, const int* in_sizes, int n_in,
                              void* d_out, int out_size, void* d_ws, size_t ws_size,
                              hipStream_t stream)
{
    const int*   context_ids = (const int*)  d_in[0];
    const int*   target_ids  = (const int*)  d_in[1];
    const int*   context_len = (const int*)  d_in[2];
    const int*   target_len  = (const int*)  d_in[3];
    const int*   target_off  = (const int*)  d_in[4];
    const float* E           = (const float*)d_in[5];
    const float* W_lin       = (const float*)d_in[6];
    const float* b_lin       = (const float*)d_in[7];
    const float* w_attn      = (const float*)d_in[8];
    const float* b_attn      = (const float*)d_in[9];
    const float* W_out       = (const float*)d_in[10];
    const float* b_out       = (const float*)d_in[11];
    float*       out         = (float*)d_out;

    float* ws   = (float*)d_ws;
    float* s_mem = ws;                       // B*L
    float* vloc  = s_mem + B_ * L_;          // B*L
    float* wbl   = vloc  + B_ * L_;          // B*L  (alpha * v_loc)
    float* vecA  = wbl   + B_ * L_;          // B*D
    float* vecB  = vecA  + B_ * D_;          // B*D
    float* attn  = vecB  + B_ * D_;          // B*D

    // vec0 = v_aspect
    k_vaspect<<<B_, 256, 0, stream>>>(target_ids, target_len, E, vecA);

    // hop-invariant content scores + location weights
    k_smem<<<(B_ * L_) / 8, 256, 0, stream>>>(context_ids, context_len, target_off,
                                              E, w_attn, s_mem, vloc);

    float* cur = vecA;
    float* nxt = vecB;
    for (int hop = 0; hop < 6; ++hop) {
        k_softmax<<<B_, 256, 0, stream>>>(cur, s_mem, vloc, w_attn, b_attn,
                                          context_len, wbl);
        k_attn<<<dim3(D_ / 256, B_), 256, 0, stream>>>(context_ids, context_len,
                                                       E, wbl, attn);
        k_lin_wmma<<<32, 256, 0, stream>>>(cur, W_lin, b_lin, attn, nxt);
        float* t = cur; cur = nxt; nxt = t;
    }

    k_logits<<<B_ * C_, 256, 0, stream>>>(cur, W_out, b_out, out);
}